// GraphTrans_40011915329894
// MI455X (gfx1250) — compile-verified
//
#include <hip/hip_runtime.h>
#include <hip/hip_bf16.h>
#include <math.h>

// Problem constants (match reference)
#define BB      4
#define NN      1500
#define KNB     30
#define NODE_F  1024
#define HH      128
#define LL      4
#define NHEADS  4
#define DHD     32
#define NNODE   (BB * NN)        // 6000
#define NEDGE   (BB * NN * KNB)  // 180000

typedef __attribute__((ext_vector_type(2))) float v2f;
typedef __attribute__((ext_vector_type(8))) float v8f;
typedef int v4i_ __attribute__((vector_size(4 * sizeof(int))));

// Full-precision fp32 matrix op: D(16x16) += A(16x4) * B(4x16)
__device__ __forceinline__ v8f wmma4(v2f a, v2f b, v8f c) {
  return __builtin_amdgcn_wmma_f32_16x16x4_f32(
      /*neg_a=*/false, a, /*neg_b=*/false, b,
      /*c_mod=*/(short)0, c, /*reuse_a=*/false, /*reuse_b=*/false);
}

// ---------------------------------------------------------------------------
// Async global->LDS 16-byte copy (CDNA5 GLOBAL_LOAD_ASYNC_TO_LDS_B128, per-lane
// addresses, tracked by ASYNCcnt, bypasses VGPRs). Falls back to a synchronous
// copy if the builtin is not declared by this toolchain.
// ---------------------------------------------------------------------------
#if defined(__has_builtin)
#if __has_builtin(__builtin_amdgcn_global_load_async_to_lds_b128)
#define HAS_ASYNC_LDS 1
#endif
#endif

__device__ __forceinline__ void cp_async16(float* lds_dst, const float* gsrc) {
#ifdef HAS_ASYNC_LDS
  __builtin_amdgcn_global_load_async_to_lds_b128(
      (__attribute__((address_space(1))) v4i_*)(unsigned long long)gsrc,
      (__attribute__((address_space(3))) v4i_*)(unsigned)(unsigned long long)lds_dst,
      0, 0);
#else
  float4 v = *(const float4*)gsrc;
  *(float4*)lds_dst = v;
#endif
}

__device__ __forceinline__ void wait_async0() {
#ifdef HAS_ASYNC_LDS
#if __has_builtin(__builtin_amdgcn_s_wait_asynccnt)
  __builtin_amdgcn_s_wait_asynccnt(0);
#else
  asm volatile("s_wait_asynccnt 0x0" ::: "memory");
#endif
#endif
}

// ---------------------------------------------------------------------------
// kNN top-K: one block per (b,i); D row staged in LDS, iterative argmin with
// tie-break on lower index (matches jax.lax.top_k ordering on -D_adj).
// ---------------------------------------------------------------------------
__global__ __launch_bounds__(128)
void k_topk(const float* __restrict__ X, const float* __restrict__ msk,
            int* __restrict__ Eidx, float* __restrict__ Dnb)
{
  __shared__ float Drow[NN];
  __shared__ float Mrow[NN];
  __shared__ float red[128];
  __shared__ int   redi[128];
  const int tid  = threadIdx.x;
  const int node = blockIdx.x;
  const int b    = node / NN;
  const float x0 = X[(size_t)node * 3 + 0];
  const float x1 = X[(size_t)node * 3 + 1];
  const float x2 = X[(size_t)node * 3 + 2];
  const float mi = msk[node];

  float lmax = -3.4e38f;
  for (int j = tid; j < NN; j += 128) {
    float dx = x0 - X[((size_t)b * NN + j) * 3 + 0];
    float dy = x1 - X[((size_t)b * NN + j) * 3 + 1];
    float dz = x2 - X[((size_t)b * NN + j) * 3 + 2];
    float m2 = mi * msk[b * NN + j];
    float d  = m2 * sqrtf(dx * dx + dy * dy + dz * dz + 1e-6f);
    Drow[j] = d;
    Mrow[j] = m2;
    lmax = fmaxf(lmax, d);
  }
  red[tid] = lmax;
  __syncthreads();
  for (int s = 64; s > 0; s >>= 1) {
    if (tid < s) red[tid] = fmaxf(red[tid], red[tid + s]);
    __syncthreads();
  }
  const float rowmax = red[0];
  __syncthreads();
  for (int j = tid; j < NN; j += 128) Drow[j] += (1.f - Mrow[j]) * rowmax;
  __syncthreads();

  for (int r = 0; r < KNB; ++r) {
    float mv = 3.4e38f; int mx = 0;
    for (int j = tid; j < NN; j += 128) {
      float v = Drow[j];
      if (v < mv) { mv = v; mx = j; }
    }
    red[tid] = mv; redi[tid] = mx;
    __syncthreads();
    for (int s = 64; s > 0; s >>= 1) {
      if (tid < s) {
        if (red[tid + s] < red[tid] ||
            (red[tid + s] == red[tid] && redi[tid + s] < redi[tid])) {
          red[tid] = red[tid + s]; redi[tid] = redi[tid + s];
        }
      }
      __syncthreads();
    }
    if (tid == 0) {
      Eidx[(size_t)node * KNB + r] = redi[0];
      Dnb [(size_t)node * KNB + r] = red[0];
      Drow[redi[0]] = 3.4e38f;
    }
    __syncthreads();
  }
}

// ---------------------------------------------------------------------------
// Edge features: 16 positional (cos/sin) + 16 RBF per edge -> feats[NEDGE][32]
// ---------------------------------------------------------------------------
__global__ __launch_bounds__(128)
void k_edge_feats(const int* __restrict__ Eidx, const float* __restrict__ Dnb,
                  float* __restrict__ feats)
{
  int e = blockIdx.x * blockDim.x + threadIdx.x;
  if (e >= NEDGE) return;
  int node = e / KNB;
  int i    = node % NN;
  float drel = (float)Eidx[e] - (float)i;
  float* f = feats + (size_t)e * 32;
#pragma unroll
  for (int p = 0; p < 8; ++p) {
    float freq = __expf((float)(2 * p) * (-9.210340371976184f / 16.f)); // -ln(1e4)/16
    float ang  = drel * freq;
    f[p]     = __cosf(ang);
    f[8 + p] = __sinf(ang);
  }
  float d = Dnb[e];
#pragma unroll
  for (int r = 0; r < 16; ++r) {
    float mu = (20.f / 15.f) * (float)r;
    float t  = (d - mu) * 0.8f;   // / sigma, sigma = 1.25
    f[16 + r] = __expf(-t * t);
  }
}

// ---------------------------------------------------------------------------
// Generic fp32 WMMA GEMM: C[M x Nld](cols n0..n0+127) = A[M x Kd] * B[Kd x Nld]
// 64x128 block tile, BK=32, 8 waves (wave32), each wave owns 1x4 16x16 tiles.
// A/B panels double-buffered in LDS via async-to-LDS DMA: WMMA on stage t
// overlaps the ASYNCcnt-tracked copy of stage t+1.  Requires Kd % 32 == 0.
// ---------------------------------------------------------------------------
__global__ __launch_bounds__(256)
void k_gemm(const float* __restrict__ A, const float* __restrict__ Bm,
            const float* __restrict__ bias, float* __restrict__ C,
            int M, int Kd, int Nld, int relu)
{
  __shared__ float As[2][64][36];   // row stride 144B (16B aligned)
  __shared__ float Bs[2][32][132];  // row stride 528B (16B aligned)
  const int tid  = threadIdx.x;
  const int wave = tid >> 5, lane = tid & 31;
  const int half = lane >> 4, lnm = lane & 15;
  const int m0   = blockIdx.x * 64;
  const int n0   = blockIdx.y * 128;
  const int msub = wave & 3;
  const int ngrp = (wave >> 2) * 4;

  v8f zero = {};
  v8f acc[4];
#pragma unroll
  for (int j = 0; j < 4; ++j) acc[j] = zero;

  // stage(buf, k0): issue async copies of the 64x32 A panel + 32x128 B panel
  auto stage = [&](int buf, int k0) {
#pragma unroll
    for (int q = 0; q < 2; ++q) {           // A: 512 16B chunks
      int idx = tid + q * 256;
      int row = idx >> 3, c4 = (idx & 7) * 4;
      int rg  = m0 + row;
      if (rg >= M) rg = M - 1;              // clamp: pad rows are never stored
      cp_async16(&As[buf][row][c4], A + (size_t)rg * Kd + k0 + c4);
    }
#pragma unroll
    for (int q = 0; q < 4; ++q) {           // B: 1024 16B chunks
      int idx = tid + q * 256;
      int row = idx >> 5, c4 = (idx & 31) * 4;
      cp_async16(&Bs[buf][row][c4], Bm + (size_t)(k0 + row) * Nld + n0 + c4);
    }
  };

  stage(0, 0);
  wait_async0();
  __syncthreads();

  int buf = 0;
  for (int k0 = 0; k0 < Kd; k0 += 32) {
    if (k0 + 32 < Kd) stage(buf ^ 1, k0 + 32);   // overlap DMA with WMMA
#pragma unroll
    for (int kk = 0; kk < 8; ++kk) {
      int kb = kk * 4 + half * 2;
      v2f a; a.x = As[buf][msub * 16 + lnm][kb]; a.y = As[buf][msub * 16 + lnm][kb + 1];
#pragma unroll
      for (int j = 0; j < 4; ++j) {
        int n = (ngrp + j) * 16 + lnm;
        v2f bv; bv.x = Bs[buf][kb][n]; bv.y = Bs[buf][kb + 1][n];
        acc[j] = wmma4(a, bv, acc[j]);
      }
    }
    wait_async0();          // next-stage DMA landed (per-wave ASYNCcnt)
    __syncthreads();        // all waves done with `buf`, all DMA visible
    buf ^= 1;
  }
#pragma unroll
  for (int j = 0; j < 4; ++j) {
    int n  = (ngrp + j) * 16 + lnm;
    float bv = bias ? bias[n0 + n] : 0.f;
#pragma unroll
    for (int r = 0; r < 8; ++r) {
      int m  = msub * 16 + half * 8 + r;
      int mg = m0 + m;
      if (mg < M) {
        float v = acc[j][r] + bv;
        if (relu) v = fmaxf(v, 0.f);
        C[(size_t)mg * Nld + n0 + n] = v;
      }
    }
  }
}

// ---------------------------------------------------------------------------
// Residual + LayerNorm (+ optional row mask); width fixed at 128, wave per row.
// ---------------------------------------------------------------------------
__global__ __launch_bounds__(128)
void k_add_ln(float* __restrict__ h, const float* __restrict__ upd,
              const float* __restrict__ g, const float* __restrict__ bta,
              const float* __restrict__ msk, int rows)
{
  const int wave = threadIdx.x >> 5, lane = threadIdx.x & 31;
  const int row  = blockIdx.x * 4 + wave;
  if (row >= rows) return;
  float x[4];
#pragma unroll
  for (int j = 0; j < 4; ++j) {
    x[j] = h[(size_t)row * HH + lane * 4 + j];
    if (upd) x[j] += upd[(size_t)row * HH + lane * 4 + j];
  }
  float s  = x[0] + x[1] + x[2] + x[3];
  float s2 = x[0]*x[0] + x[1]*x[1] + x[2]*x[2] + x[3]*x[3];
#pragma unroll
  for (int m = 16; m >= 1; m >>= 1) {       // wave32 reduction
    s  += __shfl_xor(s,  m, 32);
    s2 += __shfl_xor(s2, m, 32);
  }
  float mean = s * (1.f / HH);
  float var  = s2 * (1.f / HH) - mean * mean;
  float inv  = rsqrtf(var + 1e-6f);
  float mm   = msk ? msk[row] : 1.f;
#pragma unroll
  for (int j = 0; j < 4; ++j) {
    int c = lane * 4 + j;
    h[(size_t)row * HH + c] = (g[c] * (x[j] - mean) * inv + bta[c]) * mm;
  }
}

// ---------------------------------------------------------------------------
// Fused neighborhood attention, one node per block (128 threads, 4 waves):
//   async-gather h_EV (32x256, zero-padded) into LDS  ->  Kh,Vh = h_EV @ Wk/Wv
//   via fp32 WMMA (accumulators in VGPRs)  ->  per-head masked softmax over
//   K=30 neighbors  ->  upd = attend @ Vh.  Never materializes h_EV/Kh/Vh.
// ---------------------------------------------------------------------------
__global__ __launch_bounds__(128)
void k_attn(const float* __restrict__ hV, const float* __restrict__ hE,
            const float* __restrict__ Q,  const int* __restrict__ Eidx,
            const float* __restrict__ msk, const float* __restrict__ Wk,
            const float* __restrict__ Wv,  float* __restrict__ upd)
{
  __shared__ float EV[32][264];   // h_EV tile; later reused for [Kh | Vh]
  __shared__ float qs[HH];
  __shared__ float mj[32];
  __shared__ float lg[NHEADS][32];
  __shared__ float at[NHEADS][32];

  const int tid  = threadIdx.x;
  const int node = blockIdx.x;
  const int b    = node / NN;
  const float mi = msk[node];

  qs[tid] = Q[(size_t)node * HH + tid];
  if (tid < 32) {
    float m = 0.f;
    if (tid < KNB) {
      int nb = Eidx[(size_t)node * KNB + tid];
      m = msk[b * NN + nb];
    }
    mj[tid] = m;
  }
  // async gather: rows 0..29 = [h_E row | h_V[neighbor] row]; 16B per lane op.
  // row base EV[k] is 1056B from EV[k-1] (16B aligned), col offsets 16B-aligned.
#pragma unroll
  for (int q = 0; q < 15; ++q) {            // 30*64 = 1920 chunks / 128 threads
    int c   = tid + q * 128;
    int k   = c >> 6;                       // 64 chunks per 256-float row
    int off = (c & 63) * 4;
    const float* src;
    if (off < HH) {
      src = hE + ((size_t)node * KNB + k) * HH + off;
    } else {
      int nb = Eidx[(size_t)node * KNB + k];
      src = hV + ((size_t)b * NN + nb) * HH + (off - HH);
    }
    cp_async16(&EV[k][off], src);
  }
  for (int c = tid; c < 2 * 256; c += 128)  // zero pad rows 30,31
    EV[30 + (c >> 8)][c & 255] = 0.f;
  wait_async0();
  __syncthreads();

  const int wave = tid >> 5, lane = tid & 31;
  const int half = lane >> 4, lnm = lane & 15;
  const int msub = wave & 1;
  const int ngrp = (wave >> 1) * 4;

  v8f zero = {};
  v8f accK[4], accV[4];
#pragma unroll
  for (int j = 0; j < 4; ++j) { accK[j] = zero; accV[j] = zero; }

  for (int kk = 0; kk < 64; ++kk) {         // K = 256 in steps of 4
    int kb = kk * 4 + half * 2;
    v2f a; a.x = EV[msub * 16 + lnm][kb]; a.y = EV[msub * 16 + lnm][kb + 1];
#pragma unroll
    for (int j = 0; j < 4; ++j) {
      int n = (ngrp + j) * 16 + lnm;
      v2f bk; bk.x = Wk[(size_t)kb * HH + n]; bk.y = Wk[(size_t)(kb + 1) * HH + n];
      accK[j] = wmma4(a, bk, accK[j]);
      v2f bw; bw.x = Wv[(size_t)kb * HH + n]; bw.y = Wv[(size_t)(kb + 1) * HH + n];
      accV[j] = wmma4(a, bw, accV[j]);
    }
  }
  __syncthreads();                          // everyone done reading EV
#pragma unroll
  for (int j = 0; j < 4; ++j) {
    int n = (ngrp + j) * 16 + lnm;
#pragma unroll
    for (int r = 0; r < 8; ++r) {
      int m = msub * 16 + half * 8 + r;
      EV[m][n]       = accK[j][r];          // Kh in cols [0,128)
      EV[m][128 + n] = accV[j][r];          // Vh in cols [128,256)
    }
  }
  __syncthreads();

  if (tid < NHEADS * KNB) {
    int hh = tid / KNB, k = tid % KNB;
    float s = 0.f;
#pragma unroll
    for (int d = 0; d < DHD; ++d) s += qs[hh * DHD + d] * EV[k][hh * DHD + d];
    s *= 0.17677669529663687f;              // 1/sqrt(DH)
    float mv = mi * mj[k];
    lg[hh][k] = (mv > 0.f) ? s : -3.4e38f;
  }
  __syncthreads();
  if (tid < NHEADS) {
    int hh = tid;
    float mx = -3.4e38f;
    for (int k = 0; k < KNB; ++k) mx = fmaxf(mx, lg[hh][k]);
    float sum = 0.f;
    for (int k = 0; k < KNB; ++k) {
      float e = (lg[hh][k] > -1e38f) ? __expf(lg[hh][k] - mx) : 0.f;
      at[hh][k] = e; sum += e;
    }
    float inv = (sum > 0.f) ? 1.f / sum : 0.f;
    for (int k = 0; k < KNB; ++k) at[hh][k] *= inv * (mi * mj[k]);
  }
  __syncthreads();
  {
    int d = tid, hh = d / DHD;
    float s = 0.f;
    for (int k = 0; k < KNB; ++k) s += at[hh][k] * EV[k][128 + d];
    upd[(size_t)node * HH + d] = s;
  }
}

// ---------------------------------------------------------------------------
// Final projection: out[row] = h_V[row] . W_out + b_out
// ---------------------------------------------------------------------------
__global__ __launch_bounds__(128)
void k_out(const float* __restrict__ h, const float* __restrict__ w,
           const float* __restrict__ b, float* __restrict__ out, int rows)
{
  const int wave = threadIdx.x >> 5, lane = threadIdx.x & 31;
  const int row  = blockIdx.x * 4 + wave;
  if (row >= rows) return;
  float s = 0.f;
#pragma unroll
  for (int j = 0; j < 4; ++j)
    s += h[(size_t)row * HH + lane * 4 + j] * w[lane * 4 + j];
#pragma unroll
  for (int m = 16; m >= 1; m >>= 1) s += __shfl_xor(s, m, 32);
  if (lane == 0) out[row] = s + b[0];
}

// ---------------------------------------------------------------------------
extern "C" void kernel_launch(void* const* d_in, const int* in_sizes, int n_in,
                              void* d_out, int out_size, void* d_ws, size_t ws_size,
                              hipStream_t stream)
{
  const float* X      = (const float*)d_in[0];
  const float* V      = (const float*)d_in[1];
  const float* msk    = (const float*)d_in[2];
  const float* W_v    = (const float*)d_in[3];
  const float* b_v    = (const float*)d_in[4];
  const float* W_e    = (const float*)d_in[5];
  const float* b_e    = (const float*)d_in[6];
  const float* W_edge = (const float*)d_in[7];
  const float* ln_e_g = (const float*)d_in[8];
  const float* ln_e_b = (const float*)d_in[9];
  const float* Wq     = (const float*)d_in[10];
  const float* Wk     = (const float*)d_in[11];
  const float* Wv_a   = (const float*)d_in[12];
  const float* Wo     = (const float*)d_in[13];
  const float* ln1_g  = (const float*)d_in[14];
  const float* ln1_b  = (const float*)d_in[15];
  const float* ln2_g  = (const float*)d_in[16];
  const float* ln2_b  = (const float*)d_in[17];
  const float* W_ff1  = (const float*)d_in[18];
  const float* b_ff1  = (const float*)d_in[19];
  const float* W_ff2  = (const float*)d_in[20];
  const float* b_ff2  = (const float*)d_in[21];
  const float* W_out  = (const float*)d_in[22];
  const float* b_out  = (const float*)d_in[23];
  float* out = (float*)d_out;

  char* ws = (char*)d_ws;
  size_t off = 0;
  auto carve = [&](size_t bytes) -> char* {
    char* p = ws + off;
    off = (off + bytes + 255) & ~(size_t)255;
    return p;
  };
  int*   Eidx  = (int*)  carve(sizeof(int)   * NEDGE);
  float* Dnb   = (float*)carve(sizeof(float) * NEDGE);
  float* feats = (float*)carve(sizeof(float) * NEDGE * 32);
  float* Ebuf  = (float*)carve(sizeof(float) * (size_t)NEDGE * HH);
  float* hE    = (float*)carve(sizeof(float) * (size_t)NEDGE * HH);
  float* hV    = (float*)carve(sizeof(float) * NNODE * HH);
  float* Qb    = (float*)carve(sizeof(float) * NNODE * HH);
  float* updb  = (float*)carve(sizeof(float) * NNODE * HH);
  float* upd2  = (float*)carve(sizeof(float) * NNODE * HH);
  float* ffb   = (float*)carve(sizeof(float) * NNODE * 4 * HH);

  // Graph construction + edge embedding
  k_topk<<<NNODE, 128, 0, stream>>>(X, msk, Eidx, Dnb);
  k_edge_feats<<<(NEDGE + 127) / 128, 128, 0, stream>>>(Eidx, Dnb, feats);
  k_gemm<<<dim3((NEDGE + 63) / 64, 1), 256, 0, stream>>>(feats, W_edge, nullptr, Ebuf, NEDGE, 32, HH, 0);
  k_add_ln<<<(NEDGE + 3) / 4, 128, 0, stream>>>(Ebuf, nullptr, ln_e_g, ln_e_b, nullptr, NEDGE);
  k_gemm<<<dim3((NEDGE + 63) / 64, 1), 256, 0, stream>>>(Ebuf, W_e, b_e, hE, NEDGE, HH, HH, 0);
  k_gemm<<<dim3((NNODE + 63) / 64, 1), 256, 0, stream>>>(V, W_v, b_v, hV, NNODE, NODE_F, HH, 0);

  for (int l = 0; l < LL; ++l) {
    const float* Wq_l = Wq   + (size_t)l * HH * HH;
    const float* Wk_l = Wk   + (size_t)l * 2 * HH * HH;
    const float* Wv_l = Wv_a + (size_t)l * 2 * HH * HH;
    const float* Wo_l = Wo   + (size_t)l * HH * HH;

    k_gemm<<<dim3((NNODE + 63) / 64, 1), 256, 0, stream>>>(hV, Wq_l, nullptr, Qb, NNODE, HH, HH, 0);
    k_attn<<<NNODE, 128, 0, stream>>>(hV, hE, Qb, Eidx, msk, Wk_l, Wv_l, updb);
    k_gemm<<<dim3((NNODE + 63) / 64, 1), 256, 0, stream>>>(updb, Wo_l, nullptr, upd2, NNODE, HH, HH, 0);
    k_add_ln<<<(NNODE + 3) / 4, 128, 0, stream>>>(hV, upd2, ln1_g + l * HH, ln1_b + l * HH, nullptr, NNODE);

    k_gemm<<<dim3((NNODE + 63) / 64, 4), 256, 0, stream>>>(hV, W_ff1 + (size_t)l * HH * 4 * HH,
                                                           b_ff1 + l * 4 * HH, ffb, NNODE, HH, 4 * HH, 1);
    k_gemm<<<dim3((NNODE + 63) / 64, 1), 256, 0, stream>>>(ffb, W_ff2 + (size_t)l * 4 * HH * HH,
                                                           b_ff2 + l * HH, upd2, NNODE, 4 * HH, HH, 0);
    k_add_ln<<<(NNODE + 3) / 4, 128, 0, stream>>>(hV, upd2, ln2_g + l * HH, ln2_b + l * HH, msk, NNODE);
  }

  k_out<<<(NNODE + 3) / 4, 128, 0, stream>>>(hV, W_out, b_out, out, NNODE);
}